// kernel_generated_0_21887153341212
// MI455X (gfx1250) — compile-verified
//
#include <hip/hip_runtime.h>

typedef __attribute__((ext_vector_type(2))) float v2f;
typedef __attribute__((ext_vector_type(8))) float v8f;

#define HH     56
#define WW     56
#define KTOT   192          // 64 channels * 3 taps
#define LDSTR  194          // LDS row stride (floats): even (b64 align), %64==2 (bank spread)

__global__ __launch_bounds__(128)
void conv3tap_roll_wmma(const float* __restrict__ x,
                        const float* __restrict__ wgt,
                        float* __restrict__ out)
{
    // B panel for this output row: sB[wcol][K] = t2[c, h+kk-1, wcol], K = c*3+kk
    __shared__ float sB[64 * LDSTR];   // 49,664 bytes

    const int h    = blockIdx.x;       // output row 0..55
    const int lane = threadIdx.x;      // 0..31 (wave32)
    const int wv   = threadIdx.y;      // 0..3  : wave id == 16-wide w-tile id
    const int tid  = wv * 32 + lane;   // 0..127

    // ---- cooperative stage of B into LDS (pair-sum + 3-tap gather, zero pad) ----
    for (int idx = tid; idx < KTOT * 64; idx += 128) {
        const int K  = idx >> 6;       // 0..191  (consecutive tids -> consecutive w: coalesced)
        const int wc = idx & 63;       // 0..63 (columns 56..63 are pad)
        const int c  = K / 3;
        const int kk = K - 3 * c;
        const int hp = h + kk - 1;
        float v = 0.0f;
        if (hp >= 0 && hp < HH && wc < WW) {
            const int o = hp * WW + wc;
            v = x[c * (HH * WW) + o] + x[(c + 64) * (HH * WW) + o];
        }
        sB[wc * LDSTR + K] = v;
    }
    __syncthreads();

    // ---- WMMA fragment addressing (V_WMMA_F32_16X16X4_F32 layout) ----
    // lanes 0-15 carry K=4s,4s+1 ; lanes 16-31 carry K=4s+2,4s+3
    const int half = lane >> 4;
    const int ln   = lane & 15;
    const int w0   = wv * 16;

    const float* bP  = &sB[(w0 + ln) * LDSTR + 2 * half];   // ds_load_b64 per step
    const float* aP0 = wgt + (0  + ln) * KTOT + 2 * half;   // A rows 0..15  (global, L1-hot)
    const float* aP1 = wgt + (16 + ln) * KTOT + 2 * half;   // A rows 16..31

    v8f acc0 = {};   // D rows 0..15  of this w-tile
    v8f acc1 = {};   // D rows 16..31

    #pragma unroll 8
    for (int s = 0; s < KTOT / 4; ++s) {
        const v2f b  = *(const v2f*)(bP  + 4 * s);
        const v2f a0 = *(const v2f*)(aP0 + 4 * s);
        const v2f a1 = *(const v2f*)(aP1 + 4 * s);
        // 8-arg form: (neg_a, A, neg_b, B, c_mod, C, reuse_a, reuse_b)
        acc0 = __builtin_amdgcn_wmma_f32_16x16x4_f32(false, a0, false, b,
                                                     (short)0, acc0, false, false);
        acc1 = __builtin_amdgcn_wmma_f32_16x16x4_f32(false, a1, false, b,
                                                     (short)0, acc1, false, false);
    }

    // ---- store with roll(+1, axis=w); skip pad columns ----
    const int wcol = w0 + ln;
    if (wcol < WW) {
        const int wr = (wcol == WW - 1) ? 0 : wcol + 1;
        #pragma unroll
        for (int v = 0; v < 8; ++v) {
            const int m = v + half * 8;      // C/D layout: VGPR v, lane half selects M / M+8
            out[(m)      * (HH * WW) + h * WW + wr] = acc0[v];
            out[(m + 16) * (HH * WW) + h * WW + wr] = acc1[v];
        }
    }
}

extern "C" void kernel_launch(void* const* d_in, const int* in_sizes, int n_in,
                              void* d_out, int out_size, void* d_ws, size_t ws_size,
                              hipStream_t stream) {
    (void)in_sizes; (void)n_in; (void)d_ws; (void)ws_size; (void)out_size;
    const float* x   = (const float*)d_in[0];   // (1,128,56,56) f32
    const float* wgt = (const float*)d_in[1];   // (32,64,3)     f32
    float* out = (float*)d_out;                 // (1,32,56,56)  f32

    dim3 grid(HH);          // one block per output row
    dim3 block(32, 4);      // 4 wave32 waves: one per 16-wide w-tile
    conv3tap_roll_wmma<<<grid, block, 0, stream>>>(x, wgt, out);
}